// ChildSumTreeLSTM_34574486733552
// MI455X (gfx1250) — compile-verified
//
#include <hip/hip_runtime.h>
#include <math.h>

// ---------------------------------------------------------------------------
// Child-Sum TreeLSTM, perfect binary tree depth 17, dims 128, fp32.
// Strategy: per-level kernel (leaves -> root). Each 256-thread block (8 wave32)
// processes 16 nodes. Gate pre-activations computed with native fp32 WMMA
// (v_wmma_f32_16x16x4_f32), accumulators kept in registers, LSTM pointwise
// math applied directly on the C-fragment layout.
// ---------------------------------------------------------------------------

#define TREE_DEPTH 17
#define NNODES ((1 << TREE_DEPTH) - 1)   // 131071
#define HID 128

typedef __attribute__((ext_vector_type(2))) float v2f;
typedef __attribute__((ext_vector_type(8))) float v8f;

__device__ __forceinline__ float sigmoidf_(float v) {
    return 1.0f / (1.0f + __expf(-v));
}

// One fp32 WMMA K-step: C[16x16] += A[16x4] * B[4x16]
__device__ __forceinline__ v8f wmma_step(v2f a, v2f b, v8f c) {
    return __builtin_amdgcn_wmma_f32_16x16x4_f32(
        /*neg_a=*/false, a, /*neg_b=*/false, b,
        /*c_mod=*/(short)0, c, /*reuse_a=*/false, /*reuse_b=*/false);
}

__global__ __launch_bounds__(256) void tree_level_kernel(
    const float* __restrict__ x,
    const float* __restrict__ Wi, const float* __restrict__ bWi,
    const float* __restrict__ Ui, const float* __restrict__ bUi,
    const float* __restrict__ Wf, const float* __restrict__ bWf,
    const float* __restrict__ Uf, const float* __restrict__ bUf,
    const float* __restrict__ Wo, const float* __restrict__ bWo,
    const float* __restrict__ Uo, const float* __restrict__ bUo,
    const float* __restrict__ Wu, const float* __restrict__ bWu,
    const float* __restrict__ Uu, const float* __restrict__ bUu,
    float* __restrict__ hbuf, float* __restrict__ cbuf,
    int s, int cnt, int leaf)
{
    // LDS staging: 16-node tile of x, child-h (left/right) and their sum.
    __shared__ float Ax[16 * HID];
    __shared__ float HL[16 * HID];
    __shared__ float HR[16 * HID];
    __shared__ float HS[16 * HID];

    const int tid   = threadIdx.x;
    const int tbase = blockIdx.x * 16;

    // ------------------ load phase (zero-pad invalid rows) ------------------
    {
        const int r  = tid >> 4;         // 0..15 tile row
        const int c0 = (tid & 15) * 8;   // 8 floats per thread
        const int node  = tbase + r;
        const bool valid = node < cnt;
        const int g = s + node;
        #pragma unroll
        for (int j = 0; j < 8; ++j) {
            const int c = c0 + j;
            Ax[r * HID + c] = valid ? x[(size_t)g * HID + c] : 0.0f;
        }
        if (!leaf) {
            const int gl = 2 * g + 1, gr = 2 * g + 2;
            #pragma unroll
            for (int j = 0; j < 8; ++j) {
                const int c = c0 + j;
                const float hl = valid ? hbuf[(size_t)gl * HID + c] : 0.0f;
                const float hr = valid ? hbuf[(size_t)gr * HID + c] : 0.0f;
                HL[r * HID + c] = hl;
                HR[r * HID + c] = hr;
                HS[r * HID + c] = hl + hr;
            }
        }
    }
    __syncthreads();

    // ------------------ WMMA phase (per-wave 16-column tile) ----------------
    const int lane   = tid & 31;
    const int w      = tid >> 5;              // wave id 0..7
    const int n      = w * 16 + (lane & 15);  // output column (0..127)
    const int m      = lane & 15;             // A-matrix row this lane feeds
    const int kb     = (lane >> 4) * 2;       // per-lane K sub-offset (0 or 2)
    const int rowoff = (lane >> 4) * 8;       // C fragment base row

    // Weight rows: B[k][n] = W[n][k] since W is [out][in] row-major.
    const float* wi = Wi + (size_t)n * HID;
    const float* wf = Wf + (size_t)n * HID;
    const float* wo = Wo + (size_t)n * HID;
    const float* wu = Wu + (size_t)n * HID;
    const float* ui = Ui + (size_t)n * HID;
    const float* uo = Uo + (size_t)n * HID;
    const float* uu = Uu + (size_t)n * HID;
    const float* uf = Uf + (size_t)n * HID;

    v8f ai = {}, ao = {}, au = {}, afx = {}, afl = {}, afr = {};

    // x @ {Wi,Wo,Wu,Wf}^T : one A-fragment feeds 4 WMMAs per K-step.
    for (int k0 = 0; k0 < HID; k0 += 4) {
        const v2f a = *(const v2f*)&Ax[m * HID + k0 + kb];
        ai  = wmma_step(a, *(const v2f*)&wi[k0 + kb], ai);
        ao  = wmma_step(a, *(const v2f*)&wo[k0 + kb], ao);
        au  = wmma_step(a, *(const v2f*)&wu[k0 + kb], au);
        afx = wmma_step(a, *(const v2f*)&wf[k0 + kb], afx);
    }
    if (!leaf) {
        // h_sum @ {Ui,Uo,Uu}^T
        for (int k0 = 0; k0 < HID; k0 += 4) {
            const v2f a = *(const v2f*)&HS[m * HID + k0 + kb];
            ai = wmma_step(a, *(const v2f*)&ui[k0 + kb], ai);
            ao = wmma_step(a, *(const v2f*)&uo[k0 + kb], ao);
            au = wmma_step(a, *(const v2f*)&uu[k0 + kb], au);
        }
        // hL @ Uf^T and hR @ Uf^T (per-child forget gates)
        for (int k0 = 0; k0 < HID; k0 += 4) {
            const v2f a = *(const v2f*)&HL[m * HID + k0 + kb];
            afl = wmma_step(a, *(const v2f*)&uf[k0 + kb], afl);
        }
        for (int k0 = 0; k0 < HID; k0 += 4) {
            const v2f a = *(const v2f*)&HR[m * HID + k0 + kb];
            afr = wmma_step(a, *(const v2f*)&uf[k0 + kb], afr);
        }
    }

    // ------------- pointwise LSTM math directly on C fragments --------------
    // C/D layout: lane covers column n; VGPR j holds row (rowoff + j).
    const float bwi = bWi[n];
    const float bwo = bWo[n];
    const float bwu = bWu[n];
    const float bui = leaf ? 0.0f : bUi[n];
    const float buo = leaf ? 0.0f : bUo[n];
    const float buu = leaf ? 0.0f : bUu[n];
    const float bf  = leaf ? 0.0f : (bWf[n] + bUf[n]);

    #pragma unroll
    for (int j = 0; j < 8; ++j) {
        const int r    = rowoff + j;
        const int node = tbase + r;
        const bool valid = node < cnt;
        const int g = s + node;

        const float iv = sigmoidf_(ai[j] + bwi + bui);
        const float ov = sigmoidf_(ao[j] + bwo + buo);
        const float uv = tanhf(au[j] + bwu + buu);
        float cv = iv * uv;
        if (!leaf) {
            const float fb = afx[j] + bf;
            const float cl = valid ? cbuf[(size_t)(2 * g + 1) * HID + n] : 0.0f;
            const float cr = valid ? cbuf[(size_t)(2 * g + 2) * HID + n] : 0.0f;
            cv += sigmoidf_(fb + afl[j]) * cl + sigmoidf_(fb + afr[j]) * cr;
        }
        const float hv = ov * tanhf(cv);
        if (valid) {
            hbuf[(size_t)g * HID + n] = hv;
            cbuf[(size_t)g * HID + n] = cv;
        }
    }
}

// out[o] = bWp[o] + sum_k h_root[k] * Wp[o][k]   (tiny: 128x128)
__global__ __launch_bounds__(128) void root_proj_kernel(
    const float* __restrict__ hbuf,
    const float* __restrict__ Wp,
    const float* __restrict__ bWp,
    float* __restrict__ out)
{
    const int o = threadIdx.x;
    const float* wr = Wp + (size_t)o * HID;
    float acc = bWp[o];
    #pragma unroll 8
    for (int k = 0; k < HID; ++k) acc += hbuf[k] * wr[k];
    out[o] = acc;
}

extern "C" void kernel_launch(void* const* d_in, const int* in_sizes, int n_in,
                              void* d_out, int out_size, void* d_ws, size_t ws_size,
                              hipStream_t stream) {
    const float* x   = (const float*)d_in[0];
    // d_in[1] = edge_index (tree structure is implicit: parent = (child-1)/2)
    const float* Wi  = (const float*)d_in[2];  const float* bWi = (const float*)d_in[3];
    const float* Ui  = (const float*)d_in[4];  const float* bUi = (const float*)d_in[5];
    const float* Wf  = (const float*)d_in[6];  const float* bWf = (const float*)d_in[7];
    const float* Uf  = (const float*)d_in[8];  const float* bUf = (const float*)d_in[9];
    const float* Wo  = (const float*)d_in[10]; const float* bWo = (const float*)d_in[11];
    const float* Uo  = (const float*)d_in[12]; const float* bUo = (const float*)d_in[13];
    const float* Wu  = (const float*)d_in[14]; const float* bWu = (const float*)d_in[15];
    const float* Uu  = (const float*)d_in[16]; const float* bUu = (const float*)d_in[17];
    const float* Wp  = (const float*)d_in[18]; const float* bWp = (const float*)d_in[19];

    float* hbuf = (float*)d_ws;
    float* cbuf = hbuf + (size_t)NNODES * HID;

    for (int d = TREE_DEPTH - 1; d >= 0; --d) {
        const int s    = (1 << d) - 1;
        const int cnt  = 1 << d;
        const int leaf = (d == TREE_DEPTH - 1) ? 1 : 0;
        dim3 grid((cnt + 15) / 16);
        tree_level_kernel<<<grid, 256, 0, stream>>>(
            x, Wi, bWi, Ui, bUi, Wf, bWf, Uf, bUf,
            Wo, bWo, Uo, bUo, Wu, bWu, Uu, bUu,
            hbuf, cbuf, s, cnt, leaf);
    }
    root_proj_kernel<<<1, 128, 0, stream>>>(hbuf, Wp, bWp, (float*)d_out);
}